// MyModel_7121055777413
// MI455X (gfx1250) — compile-verified
//
#include <hip/hip_runtime.h>
#include <math.h>

// ---------------------------------------------------------------------------
// MI455X (gfx1250) implementation of the PEQ -> comp/exp -> FDN-reverb chain.
// All heavy lifting is pow-2 complex FFTs built from radix-16 Stockham stages
// whose 16-point DFT butterflies run on V_WMMA_F32_16X16X4_F32, with tiles
// staged into LDS via GLOBAL_LOAD_ASYNC_TO_LDS_B128 (ASYNCcnt).  The residual
// log2 N mod 4 = 3 for T is handled by a single register-resident radix-8
// pass (instead of three radix-2 passes) to minimize HBM traffic.
//   T  = 2^19  (PEQ circular filtering, exact)
//   NC = 2^20  (FDN convolution; T+IR_LEN=788888 <= 2^20 so linear conv exact)
// Workspace layout (requires ~640 MB of d_ws):
//   bufX  : 32ch * 2^20 complex  (256 MB)
//   bufY  : 32ch * 2^20 complex  (256 MB)
//   Hmain : (T/2+1) complex      (4 MB reserved)
//   Hc    : 2*(NC/2+1) complex   (16 MB reserved)
//   xr,e,y: 3 * B*T floats       (96 MB)
//   Afb   : 6x6 floats
// ---------------------------------------------------------------------------

typedef __attribute__((ext_vector_type(2))) float v2f;
typedef __attribute__((ext_vector_type(8))) float v8f;

#define PI_F 3.14159265358979323846f
#define SRATE 44100.0f
#define B_SZ 16
#define T_SZ 524288      // 2^19
#define NC_SZ 1048576    // 2^20

// ------------------------------ complex helpers ----------------------------
struct cpx { float re, im; };
__device__ __forceinline__ cpx cmk(float r, float i){ cpx c; c.re=r; c.im=i; return c; }
__device__ __forceinline__ cpx cadd(cpx a, cpx b){ return cmk(a.re+b.re, a.im+b.im); }
__device__ __forceinline__ cpx csub(cpx a, cpx b){ return cmk(a.re-b.re, a.im-b.im); }
__device__ __forceinline__ cpx cmul(cpx a, cpx b){ return cmk(a.re*b.re - a.im*b.im, a.re*b.im + a.im*b.re); }
__device__ __forceinline__ cpx cscale(cpx a, float s){ return cmk(a.re*s, a.im*s); }
__device__ __forceinline__ cpx cinv(cpx a){
  float d = 1.0f/(a.re*a.re + a.im*a.im + 1e-30f);
  return cmk(a.re*d, -a.im*d);
}
__device__ __forceinline__ float sigmoidf_(float x){ return 1.0f/(1.0f+__expf(-x)); }
__device__ __forceinline__ float softplusf_(float x){ return (x > 20.0f) ? x : __logf(1.0f + __expf(x)); }

// RBJ peaking-EQ biquad frequency response at digital freq w (rad/sample),
// with the reference's raw->param transform folded in. gain = 6 * g_raw dB.
__device__ __forceinline__ cpx biquad_resp(float w, float w0raw, float qraw, float graw)
{
  float f   = 20.0f * __powf(1000.0f, sigmoidf_(w0raw));
  float w0  = (2.0f*PI_F/SRATE) * f;
  float q   = softplusf_(qraw) + 0.3f;
  float gdb = 6.0f * graw;
  float A   = __powf(10.0f, gdb * 0.025f);         // 10^(g/40)
  float al  = __sinf(w0) / (2.0f*q);
  float cw  = __cosf(w0);
  float sw_, cw_;
  __sincosf(w, &sw_, &cw_);
  float zr = cw_, zi = -sw_;                        // z^-1 = e^{-iw}
  float z2r = zr*zr - zi*zi, z2i = 2.0f*zr*zi;      // z^-2
  float b0 = 1.0f + al*A, b2 = 1.0f - al*A;
  float a0 = 1.0f + al/A, a2 = 1.0f - al/A;
  float m  = -2.0f*cw;
  cpx num = cmk(b0 + m*zr + b2*z2r, m*zi + b2*z2i);
  cpx den = cmk(a0 + m*zr + a2*z2r, m*zi + a2*z2i);
  return cmul(num, cinv(den));
}

// ------------------- radix-16 Stockham stage on WMMA f32 -------------------
// One wave handles 16 consecutive Stockham columns j (a 16x16 complex tile):
//   v[r] = src[j + r*N/16] * e^{dir*2*pi*i * r*(j%Ns)/(16*Ns)}
//   y    = W16 * v    (W16[m][r] = e^{dir*2*pi*i*m*r/16})  via 4 real GEMMs
//   dst[(j/Ns)*16*Ns + (j%Ns) + r*Ns] = y[r]
// The raw tile is staged global->LDS with GLOBAL_LOAD_ASYNC_TO_LDS_B128
// (per-lane addresses, 4 x 16B per lane); the Stockham twiddle is applied
// while building the WMMA B operands.  Complex GEMM = 4 real GEMMs; each
// real 16x16x16 GEMM = 4 chained v_wmma_f32_16x16x4_f32.
__global__ void __launch_bounds__(128)
fft_r16_wmma(const float2* __restrict__ src, float2* __restrict__ dst,
             int N, int Ns, float dir)
{
  // row stride 18 float2 = 144 B keeps every (r, even-j) slot 16B aligned
  __shared__ float2 ldsv[4][16][18];

  const int lane = threadIdx.x & 31;
  const int wave = threadIdx.x >> 5;
  const long tile = (long)blockIdx.x * 4 + wave;
  const int tilesPerBatch = (N >> 4) >> 4;          // (N/16)/16
  const int batch = (int)(tile / tilesPerBatch);
  const int j0 = ((int)(tile % tilesPerBatch)) << 4;
  const int NdR = N >> 4;

  const float2* __restrict__ s = src + (long)batch * N;
  float2* __restrict__ d = dst + (long)batch * N;

  // hint next tile's first line into cache (global_prefetch_b8)
  __builtin_prefetch(&s[j0 + 16], 0, 1);

  // ---- async copy of the 16x16 complex tile into LDS (raw, no twiddle) ----
  // 128 x 16B transfers; lane issues 4. Element (r, jj): global s[j0+jj + r*N/16].
#pragma unroll
  for (int it = 0; it < 4; ++it) {
    const int task = lane + 32*it;          // 0..127
    const int r  = task >> 3;               // 0..15
    const int jj = (task & 7) << 1;         // 0,2,..,14  (b128 = 2 complex)
    const float2* gp = s + (size_t)(j0 + jj) + (size_t)r * NdR;
    unsigned laddr = (unsigned)(uintptr_t)&ldsv[wave][r][jj];
    unsigned long long gaddr = (unsigned long long)(uintptr_t)gp;
    asm volatile("global_load_async_to_lds_b128 %0, %1, off"
                 :: "v"(laddr), "v"(gaddr)
                 : "memory");
  }
  asm volatile("s_wait_asynccnt 0x0" ::: "memory");
  __syncthreads();

  // WMMA layouts (ISA 7.12.2):
  //  A 16x4 : lane<16 holds A[m][k0..k0+1] (k0=0), lanes 16-31 k0=2
  //  B 4x16 : lane<16 holds rows 0,2 at col=lane; lanes 16-31 rows 1,3
  //  C 16x16: lane<16 = rows 0..7 of col=lane; lanes 16-31 rows 8..15
  v8f cre = {};
  v8f cim = {};
  const int m  = lane & 15;
  const int hi = lane >> 4;
  const int jm = (j0 + m) % Ns;                         // column's (j % Ns)
  const float twb  = dir * 2.0f * PI_F / (float)(Ns * 16);
  const float wang = dir * 2.0f * PI_F / 16.0f;

#pragma unroll
  for (int ch = 0; ch < 4; ++ch) {
    // A operand: W16 slice, computed analytically in registers
    const int k0 = 4*ch + 2*hi;
    float sa0, ca0, sa1, ca1;
    __sincosf(wang * (float)(m * k0),       &sa0, &ca0);
    __sincosf(wang * (float)(m * (k0 + 1)), &sa1, &ca1);
    v2f a_re  = {  ca0,  ca1 };
    v2f a_im  = {  sa0,  sa1 };
    v2f a_imn = { -sa0, -sa1 };

    // B operand: rows r0,r1 of the tile at column m, twiddled on the fly
    const int r0 = 4*ch + hi;
    const int r1 = r0 + 2;
    float2 v0 = ldsv[wave][r0][m];
    float2 v1 = ldsv[wave][r1][m];
    float st0, ct0, st1, ct1;
    __sincosf(twb * (float)(r0 * jm), &st0, &ct0);
    __sincosf(twb * (float)(r1 * jm), &st1, &ct1);
    v2f b_re = { v0.x*ct0 - v0.y*st0, v1.x*ct1 - v1.y*st1 };
    v2f b_im = { v0.x*st0 + v0.y*ct0, v1.x*st1 + v1.y*ct1 };

    cre = __builtin_amdgcn_wmma_f32_16x16x4_f32(false, a_re,  false, b_re, (short)0, cre, false, false);
    cre = __builtin_amdgcn_wmma_f32_16x16x4_f32(false, a_imn, false, b_im, (short)0, cre, false, false);
    cim = __builtin_amdgcn_wmma_f32_16x16x4_f32(false, a_re,  false, b_im, (short)0, cim, false, false);
    cim = __builtin_amdgcn_wmma_f32_16x16x4_f32(false, a_im,  false, b_re, (short)0, cim, false, false);
  }

  // Scatter to Stockham destination.
  const int j = j0 + m;
  const long obase = (long)(j / Ns) * (long)(Ns * 16) + (long)(j % Ns);
#pragma unroll
  for (int v = 0; v < 8; ++v) {
    int r = hi*8 + v;
    float2 o;
    o.x = cre[v];
    o.y = cim[v];
    d[obase + (long)r * Ns] = o;
  }
}

// ----------------------------- radix-2 stage -------------------------------
__global__ void fft_r2(const float2* __restrict__ src, float2* __restrict__ dst,
                       int N, int Ns, float dir, int nbatch)
{
  long i = (long)blockIdx.x * blockDim.x + threadIdx.x;
  const long half = N >> 1;
  const long total = (long)nbatch * half;
  if (i >= total) return;
  const int  batch = (int)(i / half);
  const long j = i % half;
  const float2* s = src + (long)batch * N;
  float2* d = dst + (long)batch * N;
  float2 a = s[j];
  float2 b = s[j + half];
  float ang = dir * 2.0f * PI_F * (float)(j % Ns) / (float)(Ns * 2);
  float sn, cs;
  __sincosf(ang, &sn, &cs);
  float2 bt;
  bt.x = b.x*cs - b.y*sn;
  bt.y = b.x*sn + b.y*cs;
  const long ob = (j / Ns) * (long)(Ns * 2) + (j % Ns);
  d[ob]      = make_float2(a.x + bt.x, a.y + bt.y);
  d[ob + Ns] = make_float2(a.x - bt.x, a.y - bt.y);
}

// ----------------------------- radix-8 stage -------------------------------
// Register-resident 8-point DFT (DIF: e/o split + two DFT4s), natural order.
// Replaces three radix-2 global passes with one (2 fewer full-array passes).
__global__ void fft_r8(const float2* __restrict__ src, float2* __restrict__ dst,
                       int N, int Ns, float dir, int nbatch)
{
  long i = (long)blockIdx.x * blockDim.x + threadIdx.x;
  const long nb8 = N >> 3;
  const long total = (long)nbatch * nb8;
  if (i >= total) return;
  const int  batch = (int)(i / nb8);
  const long j = i % nb8;
  const float2* s = src + (long)batch * N;
  float2* d = dst + (long)batch * N;

  // load + Stockham twiddle
  cpx v[8];
  const float twb = dir * 2.0f * PI_F * (float)(j % Ns) / (float)(Ns * 8);
#pragma unroll
  for (int r = 0; r < 8; ++r) {
    float2 u = s[j + (long)r * nb8];
    if (r == 0) {
      v[0] = cmk(u.x, u.y);
    } else {
      float sn, cs;
      __sincosf(twb * (float)r, &sn, &cs);
      v[r] = cmk(u.x*cs - u.y*sn, u.x*sn + u.y*cs);
    }
  }

  // DFT8: e_r = v_r + v_{r+4};  o_r = (v_r - v_{r+4}) * W8^r
  const float s2 = 0.70710678118654752f;
  cpx e0 = cadd(v[0], v[4]), e1 = cadd(v[1], v[5]);
  cpx e2 = cadd(v[2], v[6]), e3 = cadd(v[3], v[7]);
  cpx d0 = csub(v[0], v[4]), d1 = csub(v[1], v[5]);
  cpx d2 = csub(v[2], v[6]), d3 = csub(v[3], v[7]);
  cpx o0 = d0;
  cpx o1 = cmul(d1, cmk( s2, dir*s2));              // W8^1
  cpx o2 = cmk(-dir*d2.im, dir*d2.re);              // W8^2 = i*dir
  cpx o3 = cmul(d3, cmk(-s2, dir*s2));              // W8^3

  // DFT4 (natural order): X0..X3
  cpx E[4], O[4];
  {
    cpx t0 = cadd(e0, e2), t1 = csub(e0, e2), t2 = cadd(e1, e3), t3 = csub(e1, e3);
    cpx jt3 = cmk(-dir*t3.im, dir*t3.re);
    E[0] = cadd(t0, t2); E[2] = csub(t0, t2);
    E[1] = cadd(t1, jt3); E[3] = csub(t1, jt3);
  }
  {
    cpx t0 = cadd(o0, o2), t1 = csub(o0, o2), t2 = cadd(o1, o3), t3 = csub(o1, o3);
    cpx jt3 = cmk(-dir*t3.im, dir*t3.re);
    O[0] = cadd(t0, t2); O[2] = csub(t0, t2);
    O[1] = cadd(t1, jt3); O[3] = csub(t1, jt3);
  }

  // X[2k] = E[k], X[2k+1] = O[k]
  const long ob = (j / Ns) * (long)(Ns * 8) + (j % Ns);
#pragma unroll
  for (int k = 0; k < 4; ++k) {
    d[ob + (long)(2*k    ) * Ns] = make_float2(E[k].re, E[k].im);
    d[ob + (long)(2*k + 1) * Ns] = make_float2(O[k].re, O[k].im);
  }
}

// --------------------------- small / pointwise -----------------------------
__global__ void qr6_kernel(const float* __restrict__ U, float* __restrict__ Q)
{
  if (threadIdx.x != 0 || blockIdx.x != 0) return;
  float q[6][6];
  for (int c = 0; c < 6; ++c) {
    float v[6];
    for (int r = 0; r < 6; ++r) v[r] = U[r*6 + c];
    for (int k = 0; k < c; ++k) {
      float dot = 0.0f;
      for (int r = 0; r < 6; ++r) dot += q[r][k] * v[r];
      for (int r = 0; r < 6; ++r) v[r] -= dot * q[r][k];
    }
    float n = 0.0f;
    for (int r = 0; r < 6; ++r) n += v[r]*v[r];
    float inv = rsqrtf(n + 1e-20f);
    for (int r = 0; r < 6; ++r) q[r][c] = v[r]*inv;
  }
  for (int r = 0; r < 6; ++r)
    for (int c = 0; c < 6; ++c)
      Q[r*6 + c] = q[r][c];
}

__global__ void peq_H_kernel(const float* __restrict__ w0r, const float* __restrict__ qr,
                             const float* __restrict__ gr, float2* __restrict__ H,
                             int F, int Nfft)
{
  int f = blockIdx.x * blockDim.x + threadIdx.x;
  if (f >= F) return;
  float w = 2.0f * PI_F * (float)f / (float)Nfft;
  cpx h = cmk(1.0f, 0.0f);
  for (int k = 0; k < 6; ++k) h = cmul(h, biquad_resp(w, w0r[k], qr[k], gr[k]));
  H[f] = make_float2(h.re, h.im);
}

__global__ void pack_real_kernel(const float* __restrict__ x, float2* __restrict__ c, long n)
{
  long i = (long)blockIdx.x * blockDim.x + threadIdx.x;
  if (i < n) c[i] = make_float2(x[i], 0.0f);
}

__global__ void pack_pad_kernel(const float* __restrict__ y, float2* __restrict__ c,
                                int T, int NC, int nb)
{
  long i = (long)blockIdx.x * blockDim.x + threadIdx.x;
  long tot = (long)nb * NC;
  if (i >= tot) return;
  int b = (int)(i / NC);
  int t = (int)(i % NC);
  float v = (t < T) ? y[(long)b*T + t] : 0.0f;
  c[i] = make_float2(v, 0.0f);
}

// X[b,f] *= Hermitian-extended H[f] * scale   (full-length complex spectrum)
__global__ void mulH_kernel(float2* __restrict__ X, const float2* __restrict__ H,
                            int N, int nb, float scale)
{
  long i = (long)blockIdx.x * blockDim.x + threadIdx.x;
  long tot = (long)nb * N;
  if (i >= tot) return;
  int f = (int)(i % N);
  int half = N >> 1;
  int fh = (f <= half) ? f : N - f;
  float2 h = H[fh];
  if (f > half) h.y = -h.y;
  float2 v = X[i];
  X[i] = make_float2((v.x*h.x - v.y*h.y)*scale, (v.x*h.y + v.y*h.x)*scale);
}

__global__ void env_kernel(const float2* __restrict__ xc, float* __restrict__ xr,
                           float* __restrict__ e, long n)
{
  long i = (long)blockIdx.x * blockDim.x + threadIdx.x;
  if (i >= n) return;
  float v = xc[i].x;
  xr[i] = v;
  e[i] = 0.5f * __logf(v*v + 1e-7f);
}

// one-pole attack/release envelope + comp/exp gain (sequential per batch)
__global__ void scan_kernel(const float* __restrict__ e, const float* __restrict__ xr,
                            float* __restrict__ y, int T, int B,
                            const float* att, const float* rel,
                            const float* crr, const float* errw,
                            const float* cth, const float* eth)
{
  int b = blockIdx.x * blockDim.x + threadIdx.x;
  if (b >= B) return;
  float a_att = sigmoidf_(att[0]);
  float a_rel = sigmoidf_(rel[0]);
  float cr = 1.0f + softplusf_(crr[0]);
  float er = 1.0f + softplusf_(errw[0]);
  float comp_th = cth[0], exp_th = eth[0];
  float k1 = -(1.0f - 1.0f/cr);
  float k2 = -(er - 1.0f);
  const float* eb = e  + (long)b * T;
  const float* xb = xr + (long)b * T;
  float* yb = y + (long)b * T;
  float z = eb[0];
  for (int t = 0; t < T; ++t) {
    float et = eb[t];
    float a = (et > z) ? a_att : a_rel;
    z = a*z + (1.0f - a)*et;
    float lg = k1*fmaxf(z - comp_th, 0.0f) + k2*fmaxf(exp_th - z, 0.0f);
    yb[t] = xb[t] * __expf(lg);
  }
}

// FDN transfer function on the NC grid: per frequency solve (I - D A) S = D B,
// Hc[c,f] = sum_i sum_n c[c,n] S[n,i]   (the two input channels are identical)
__global__ void fdn_spec_kernel(const float* __restrict__ Afb, const float* __restrict__ bmat,
                                const float* __restrict__ cmat,
                                const float* __restrict__ fw0, const float* __restrict__ fq,
                                const float* __restrict__ fg,
                                float2* __restrict__ Hc, int F2, int NC)
{
  int f = blockIdx.x * blockDim.x + threadIdx.x;
  if (f >= F2) return;
  const float delays[6] = {1009.0f, 1171.0f, 1307.0f, 1459.0f, 1597.0f, 1747.0f};
  float w = 2.0f * PI_F * (float)f / (float)NC;

  cpx D[6];
  for (int n = 0; n < 6; ++n) {
    cpx g = cmk(1.0f, 0.0f);
    for (int k = 0; k < 4; ++k)
      g = cmul(g, biquad_resp(w, fw0[n*4+k], fq[n*4+k], fg[n*4+k]));
    float sn, cs;
    __sincosf(-delays[n] * w, &sn, &cs);
    D[n] = cmul(g, cmk(cs, sn));
  }

  cpx M[6][6], R[6][2];
  for (int n = 0; n < 6; ++n) {
    for (int mm = 0; mm < 6; ++mm) {
      cpx t = cscale(D[n], Afb[n*6 + mm]);
      M[n][mm] = cmk(((n == mm) ? 1.0f : 0.0f) - t.re, -t.im);
    }
    R[n][0] = cscale(D[n], bmat[n*2 + 0]);
    R[n][1] = cscale(D[n], bmat[n*2 + 1]);
  }

  // Gaussian elimination with partial pivoting
  for (int col = 0; col < 6; ++col) {
    int p = col;
    float best = M[col][col].re*M[col][col].re + M[col][col].im*M[col][col].im;
    for (int r = col+1; r < 6; ++r) {
      float m2 = M[r][col].re*M[r][col].re + M[r][col].im*M[r][col].im;
      if (m2 > best) { best = m2; p = r; }
    }
    if (p != col) {
      for (int mm = col; mm < 6; ++mm) { cpx t = M[col][mm]; M[col][mm] = M[p][mm]; M[p][mm] = t; }
      for (int i = 0; i < 2; ++i)      { cpx t = R[col][i];  R[col][i]  = R[p][i];  R[p][i]  = t; }
    }
    cpx inv = cinv(M[col][col]);
    for (int r = col+1; r < 6; ++r) {
      cpx fac = cmul(M[r][col], inv);
      for (int mm = col; mm < 6; ++mm) M[r][mm] = csub(M[r][mm], cmul(fac, M[col][mm]));
      R[r][0] = csub(R[r][0], cmul(fac, R[col][0]));
      R[r][1] = csub(R[r][1], cmul(fac, R[col][1]));
    }
  }
  cpx S[6][2];
  for (int r = 5; r >= 0; --r) {
    cpx inv = cinv(M[r][r]);
    for (int i = 0; i < 2; ++i) {
      cpx acc = R[r][i];
      for (int mm = r+1; mm < 6; ++mm) acc = csub(acc, cmul(M[r][mm], S[mm][i]));
      S[r][i] = cmul(acc, inv);
    }
  }
  for (int c = 0; c < 2; ++c) {
    cpx h = cmk(0.0f, 0.0f);
    for (int i = 0; i < 2; ++i)
      for (int n = 0; n < 6; ++n)
        h = cadd(h, cscale(S[n][i], cmat[c*6 + n]));
    Hc[(long)c * F2 + f] = make_float2(h.re, h.im);
  }
}

// Z[b,c,f] = Xs[b,f] * Hermitian(Hc[c])[f] * scale
__global__ void fdn_mul_kernel(const float2* __restrict__ Xs, const float2* __restrict__ Hc,
                               float2* __restrict__ Z, int NC, int F2, int B, float scale)
{
  long i = (long)blockIdx.x * blockDim.x + threadIdx.x;
  long tot = (long)B * NC;
  if (i >= tot) return;
  int b = (int)(i / NC);
  int f = (int)(i % NC);
  int half = NC >> 1;
  int fh = (f <= half) ? f : NC - f;
  float2 X = Xs[i];
  for (int c = 0; c < 2; ++c) {
    float2 h = Hc[(long)c * F2 + fh];
    if (f > half) h.y = -h.y;
    Z[((long)(b*2 + c)) * NC + f] =
        make_float2((X.x*h.x - X.y*h.y)*scale, (X.x*h.y + X.y*h.x)*scale);
  }
}

// out[b,c,t] = Re(conv[b,c,t]) + y[b,t] * pan_amp[c]
__global__ void final_kernel(const float2* __restrict__ Zt, const float* __restrict__ y,
                             const float* __restrict__ panraw, float* __restrict__ out,
                             int T, int NC, int B)
{
  long i = (long)blockIdx.x * blockDim.x + threadIdx.x;
  long tot = (long)B * 2 * T;
  if (i >= tot) return;
  int t  = (int)(i % T);
  int bc = (int)(i / T);
  int c  = bc & 1;
  int b  = bc >> 1;
  float ang = sigmoidf_(panraw[0]) * (PI_F * 0.5f);
  float amp = ((c == 0) ? __cosf(ang) : __sinf(ang)) * 1.41421356237f;
  float conv = Zt[(long)bc * NC + t].x;
  out[i] = conv + y[(long)b*T + t] * amp;
}

// ------------------------------ host helpers -------------------------------
static float2* run_fft(float2* a, float2* b, int N, int nb, float dir, hipStream_t st)
{
  int l2 = 0; while ((1 << l2) < N) ++l2;
  int n16 = l2 / 4;
  int rem = l2 & 3;                 // 0..3 residual log2
  float2* cur = a;
  float2* alt = b;
  int Ns = 1;
  for (int i = 0; i < n16; ++i) {
    long tiles = (long)nb * (N / 256);
    fft_r16_wmma<<<(unsigned)(tiles / 4), 128, 0, st>>>(cur, alt, N, Ns, dir);
    Ns *= 16;
    float2* t = cur; cur = alt; alt = t;
  }
  if (rem == 3) {
    long total = (long)nb * (N / 8);
    fft_r8<<<(unsigned)((total + 255) / 256), 256, 0, st>>>(cur, alt, N, Ns, dir, nb);
    Ns *= 8;
    float2* t = cur; cur = alt; alt = t;
  } else {
    for (int i = 0; i < rem; ++i) {
      long total = (long)nb * (N / 2);
      fft_r2<<<(unsigned)((total + 255) / 256), 256, 0, st>>>(cur, alt, N, Ns, dir, nb);
      Ns *= 2;
      float2* t = cur; cur = alt; alt = t;
    }
  }
  return cur;
}

extern "C" void kernel_launch(void* const* d_in, const int* in_sizes, int n_in,
                              void* d_out, int out_size, void* d_ws, size_t ws_size,
                              hipStream_t stream)
{
  (void)in_sizes; (void)n_in; (void)out_size; (void)ws_size;
  const float* x    = (const float*)d_in[0];
  const float* pw0  = (const float*)d_in[1];
  const float* pq   = (const float*)d_in[2];
  const float* pg   = (const float*)d_in[3];
  const float* cth  = (const float*)d_in[4];
  const float* eth  = (const float*)d_in[5];
  const float* crr  = (const float*)d_in[6];
  const float* errw = (const float*)d_in[7];
  const float* att  = (const float*)d_in[8];
  const float* rel  = (const float*)d_in[9];
  const float* U    = (const float*)d_in[10];
  const float* bmat = (const float*)d_in[11];
  const float* cmat = (const float*)d_in[12];
  const float* fw0  = (const float*)d_in[13];
  const float* fq   = (const float*)d_in[14];
  const float* fg   = (const float*)d_in[15];
  const float* pan  = (const float*)d_in[16];

  const int B  = B_SZ;
  const int T  = T_SZ;
  const int NC = NC_SZ;
  const int F1 = T/2 + 1;
  const int F2 = NC/2 + 1;

  char* w = (char*)d_ws;
  const size_t bigbuf = (size_t)32 * NC * sizeof(float2);   // 256 MB each
  float2* bufX  = (float2*)w;
  float2* bufY  = (float2*)(w + bigbuf);
  float2* Hmain = (float2*)(w + 2*bigbuf);
  float2* Hc    = (float2*)(w + 2*bigbuf + ((size_t)4  << 20));
  float*  xr    = (float*) (w + 2*bigbuf + ((size_t)20 << 20));
  float*  ebuf  = xr   + (size_t)B*T;
  float*  ybuf  = ebuf + (size_t)B*T;
  float*  Afb   = ybuf + (size_t)B*T;

  // ---- parameter precompute ----
  qr6_kernel<<<1, 32, 0, stream>>>(U, Afb);
  peq_H_kernel<<<(F1 + 255)/256, 256, 0, stream>>>(pw0, pq, pg, Hmain, F1, T);
  fdn_spec_kernel<<<(F2 + 127)/128, 128, 0, stream>>>(Afb, bmat, cmat, fw0, fq, fg, Hc, F2, NC);

  // ---- stage 1: PEQ via length-T circular FFT filtering ----
  pack_real_kernel<<<(unsigned)(((long)B*T + 255)/256), 256, 0, stream>>>(x, bufX, (long)B*T);
  float2* cur = run_fft(bufX, bufY, T, B, -1.0f, stream);           // forward
  mulH_kernel<<<(unsigned)(((long)B*T + 255)/256), 256, 0, stream>>>(cur, Hmain, T, B, 1.0f/(float)T);
  float2* alt = (cur == bufX) ? bufY : bufX;
  cur = run_fft(cur, alt, T, B, +1.0f, stream);                     // inverse
  env_kernel<<<(unsigned)(((long)B*T + 255)/256), 256, 0, stream>>>(cur, xr, ebuf, (long)B*T);

  // ---- stage 2: compressor/expander (sequential scan per batch) ----
  scan_kernel<<<1, 32, 0, stream>>>(ebuf, xr, ybuf, T, B, att, rel, crr, errw, cth, eth);

  // ---- stage 3: FDN reverb by frequency-domain convolution at NC=2^20 ----
  pack_pad_kernel<<<(unsigned)(((long)B*NC + 255)/256), 256, 0, stream>>>(ybuf, bufX, T, NC, B);
  cur = run_fft(bufX, bufY, NC, B, -1.0f, stream);                  // Xs (16 ch)
  alt = (cur == bufX) ? bufY : bufX;
  fdn_mul_kernel<<<(unsigned)(((long)B*NC + 255)/256), 256, 0, stream>>>(cur, Hc, alt, NC, F2, B, 1.0f/(float)NC);
  cur = alt;
  alt = (cur == bufX) ? bufY : bufX;
  cur = run_fft(cur, alt, NC, 2*B, +1.0f, stream);                  // inverse (32 ch)

  // ---- pan + sum ----
  final_kernel<<<(unsigned)(((long)B*2*T + 255)/256), 256, 0, stream>>>(cur, ybuf, pan, (float*)d_out, T, NC, B);
}